// EwaldSummation_15178414424347
// MI455X (gfx1250) — compile-verified
//
#include <hip/hip_runtime.h>
#include <math.h>

// ---------------- constants (match reference) ----------------
#define ALPHA      0.4f
#define NMAX       7
#define NK         3375              // (2*NMAX+1)^3
#define KPAD       3376              // 211 * 16 (padded for 16-wide WMMA tiles)
#define KTILES     211
#define CUTOFF_R   13.14130443f      // sqrt(ln(1e12))/ALPHA
#define KCUT2      17.68385333f      // (2*ALPHA*sqrt(ln(1e12)))^2
#define INV4A2     0.390625f         // 1/(16*ALPHA^2) with ALPHA=0.4
#define CONV_FACT  14.39964548f      // 1e10*e/(4*pi*eps0)
#define SELF_COEF  (-0.2256758334f)  // -ALPHA/sqrt(pi)
#define PI_F       3.14159265358979f

typedef __attribute__((ext_vector_type(2))) float v2f;
typedef __attribute__((ext_vector_type(8))) float v8f;

// ---------------- zero the per-atom real-space accumulator ----------------
__global__ void ewald_zero_kernel(float* buf, int n) {
    int i = blockIdx.x * blockDim.x + threadIdx.x;
    if (i < n) buf[i] = 0.0f;
}

// ---------------- per-batch packed k-table + prefactor ----------------
// bpk layout per batch: [half][k][2] floats:
//   half 0 -> (kx, ky)   (B-matrix rows K=0,1)
//   half 1 -> (kz, w )   (B-matrix rows K=2,3; row 3 is multiplied by A's
//                         zero K=3 column, so stashing w there is free)
__global__ __launch_bounds__(256) void ewald_ksetup_kernel(
    const float* __restrict__ cell,   // [B,3,3] row-major
    float* __restrict__ bpk, float* __restrict__ pref)
{
    const int b = blockIdx.x;
    const float* c = cell + b * 9;
    const float c00 = c[0], c01 = c[1], c02 = c[2];
    const float c10 = c[3], c11 = c[4], c12 = c[5];
    const float c20 = c[6], c21 = c[7], c22 = c[8];

    // recip[i][j] = 2*pi * inv(cell)^T [i][j] = 2*pi * cof[i][j] / det
    const float f00 =  (c11 * c22 - c12 * c21);
    const float f01 = -(c10 * c22 - c12 * c20);
    const float f02 =  (c10 * c21 - c11 * c20);
    const float f10 = -(c01 * c22 - c02 * c21);
    const float f11 =  (c00 * c22 - c02 * c20);
    const float f12 = -(c00 * c21 - c01 * c20);
    const float f20 =  (c01 * c12 - c02 * c11);
    const float f21 = -(c00 * c12 - c02 * c10);
    const float f22 =  (c00 * c11 - c01 * c10);
    const float det = c00 * f00 + c01 * f01 + c02 * f02;   // = volume
    const float s   = (2.0f * PI_F) / det;
    const float r00 = s * f00, r01 = s * f01, r02 = s * f02;
    const float r10 = s * f10, r11 = s * f11, r12 = s * f12;
    const float r20 = s * f20, r21 = s * f21, r22 = s * f22;

    if (threadIdx.x == 0) pref[b] = 1.0f / (2.0f * det * PI_F);

    float* slab = bpk + (size_t)b * (2 * KPAD * 2);
    for (int k = threadIdx.x; k < KPAD; k += blockDim.x) {
        float kx = 0.0f, ky = 0.0f, kz = 0.0f, w = 0.0f;
        if (k < NK) {
            const int ix = k / 225;            // meshgrid 'ij': nx slowest
            const int iy = (k / 15) % 15;
            const int iz = k % 15;
            const float nx = (float)(ix - NMAX);
            const float ny = (float)(iy - NMAX);
            const float nz = (float)(iz - NMAX);
            kx = r00 * nx + r01 * ny + r02 * nz;
            ky = r10 * nx + r11 * ny + r12 * nz;
            kz = r20 * nx + r21 * ny + r22 * nz;
            const float ksq = kx * kx + ky * ky + kz * kz;
            if (ksq > 0.0f && ksq <= KCUT2)
                w = __expf(-ksq * INV4A2) / ksq;
        }
        slab[k * 2 + 0] = kx;
        slab[k * 2 + 1] = ky;
        slab[(KPAD + k) * 2 + 0] = kz;
        slab[(KPAD + k) * 2 + 1] = w;
    }
}

// ---------------- real-space: q_i q_j erfc(a r)/r scatter-add ----------------
__global__ void ewald_real_kernel(
    const float* __restrict__ dist, const int* __restrict__ eidx,
    const float* __restrict__ q, float* __restrict__ atom_real, int E)
{
    int e = blockIdx.x * blockDim.x + threadIdx.x;
    if (e >= E) return;
    float r = dist[e];
    if (r < CUTOFF_R) {
        int i = eidx[2 * e];
        int j = eidx[2 * e + 1];
        float c = q[i] * q[j] * erfcf(ALPHA * r) / r;
        atomicAdd(&atom_real[i], c);
    }
}

// ---------------- reciprocal-space via V_WMMA_F32_16X16X4_F32 ----------------
// Per block: async-stage this batch's 54KB packed k-table into LDS
// (global_load_async_to_lds_b128, ASYNCcnt), then each wave computes
// 16 atoms x 16 k-vectors of k.r per WMMA over 211 k-tiles, all operands
// from LDS. A (16x4 f32): lanes 0-15 hold K=0,1 -> (x,y); lanes 16-31 hold
// K=2,3 -> (z,0). D (16x16 f32): lane L, VGPR v -> M = v + 8*(L>>4), N = L&15.
__global__ __launch_bounds__(256) void ewald_recip_kernel(
    const float* __restrict__ pos, const float* __restrict__ q,
    const float* __restrict__ bpk, const float* __restrict__ pref,
    float* __restrict__ atom_recip, int N)
{
    __shared__ __align__(16) float s_bpk[2 * KPAD * 2];   // 54,016 bytes

    const int b    = blockIdx.y;
    const int wave = threadIdx.x >> 5;
    const int lane = threadIdx.x & 31;
    const int m    = lane & 15;          // A row / B,D column within tile
    const int half = lane >> 4;
    const int base = (blockIdx.x * 8 + wave) * 16;   // 16-atom tile base

    // ---- async copy packed k-table (batch slice) into LDS ----
    {
        const float* gsrc = bpk + (size_t)b * (2 * KPAD * 2);
        const uint32_t lds_base = (uint32_t)(uintptr_t)(&s_bpk[0]);
        for (int t = threadIdx.x; t < KPAD; t += 256) {     // KPAD x 16B = 54,016B
            uint32_t lds  = lds_base + t * 16;
            uint32_t voff = t * 16;
            asm volatile("global_load_async_to_lds_b128 %0, %1, %2"
                         :: "v"(lds), "v"(voff), "s"(gsrc) : "memory");
        }
        asm volatile("s_wait_asynccnt 0x0" ::: "memory");
        __syncthreads();
    }

    // ---- loop-invariant A operand: 16 atoms x (x,y,z,0) ----
    const int atom = base + m;
    const float* p = pos + (size_t)(b * N + atom) * 3;
    const float px = p[0], py = p[1], pz = p[2];
    v2f a;
    a.x = half ? pz : px;    // K=2 : K=0
    a.y = half ? 0.0f : py;  // K=3 : K=1  (zero kills the w row of B)

    const float* bcol = s_bpk + (size_t)half * (KPAD * 2);
    const float* wrow = s_bpk + (size_t)(KPAD * 2);

    float acc[8];
#pragma unroll
    for (int v = 0; v < 8; ++v) acc[v] = 0.0f;

#pragma unroll 2
    for (int t = 0; t < KTILES; ++t) {
        const int col = t * 16 + m;
        const v2f bm     = *(const v2f*)(bcol + col * 2);   // ds_load_b64
        const float wcol = wrow[col * 2 + 1];               // ds_load_b32

        v8f c = {};
        v8f d = __builtin_amdgcn_wmma_f32_16x16x4_f32(
            false, a, false, bm, (short)0, c, false, false);

#pragma unroll
        for (int v = 0; v < 8; ++v) {
            float sn, cs;
            __sincosf(d[v], &sn, &cs);
            acc[v] = fmaf(wcol, fmaf(sn, sn, cs * cs), acc[v]);
        }
    }

    // sum over the 16 columns held by the 16 lanes of each half
#pragma unroll
    for (int v = 0; v < 8; ++v) {
        float r = acc[v];
        r += __shfl_xor(r, 1, 32);
        r += __shfl_xor(r, 2, 32);
        r += __shfl_xor(r, 4, 32);
        r += __shfl_xor(r, 8, 32);
        acc[v] = r;
    }

    if (m == 0) {
        const float pf = pref[b];
#pragma unroll
        for (int v = 0; v < 8; ++v) {
            const int am = base + half * 8 + v;     // row M = v + 8*half
            const float qa = q[b * N + am];
            atom_recip[b * N + am] = pf * qa * qa * acc[v] * CONV_FACT;
        }
    }
}

// ---------------- per-image reduction: real + recip + self ----------------
__global__ __launch_bounds__(256) void ewald_reduce_kernel(
    const float* __restrict__ q, const float* __restrict__ atom_real,
    const float* __restrict__ atom_recip, float* __restrict__ out, int N)
{
    const int b = blockIdx.x;
    float sum = 0.0f;
    for (int a = threadIdx.x; a < N; a += 256) {
        const int g = b * N + a;
        const float qa = q[g];
        sum += atom_real[g] * (0.5f * CONV_FACT)
             + atom_recip[g]
             + SELF_COEF * qa * qa * CONV_FACT;
    }
    __shared__ float red[256];
    red[threadIdx.x] = sum;
    __syncthreads();
    for (int s = 128; s > 0; s >>= 1) {
        if (threadIdx.x < s) red[threadIdx.x] += red[threadIdx.x + s];
        __syncthreads();
    }
    if (threadIdx.x == 0) out[b] = red[0];
}

// ---------------- launcher ----------------
extern "C" void kernel_launch(void* const* d_in, const int* in_sizes, int n_in,
                              void* d_out, int out_size, void* d_ws, size_t ws_size,
                              hipStream_t stream)
{
    const float* edge_dist = (const float*)d_in[0];
    const int*   edge_idx  = (const int*)  d_in[1];
    const float* charge    = (const float*)d_in[2];
    const float* cell      = (const float*)d_in[3];
    // d_in[4] = n_atoms (unused: equal-size molecules, like the reference path)
    const float* positions = (const float*)d_in[5];
    // d_in[6] = image_idx (atoms are contiguous per image)

    const int E       = in_sizes[0];
    const int n_total = in_sizes[2];
    const int B       = in_sizes[3] / 9;
    const int N       = n_total / B;

    // workspace carve-up (floats)
    float* ws         = (float*)d_ws;
    float* bpk        = ws;                               // B * 2*KPAD*2
    float* pref       = bpk + (size_t)B * 2 * KPAD * 2;   // B
    float* atom_real  = pref + B;                         // n_total
    float* atom_recip = atom_real + n_total;              // n_total

    ewald_zero_kernel<<<(n_total + 255) / 256, 256, 0, stream>>>(atom_real, n_total);

    ewald_ksetup_kernel<<<B, 256, 0, stream>>>(cell, bpk, pref);

    ewald_real_kernel<<<(E + 255) / 256, 256, 0, stream>>>(
        edge_dist, edge_idx, charge, atom_real, E);

    // one wave per 16-atom tile: N/16 tiles per batch, 8 waves per block
    dim3 rgrid((N / 16 + 7) / 8, B);
    ewald_recip_kernel<<<rgrid, 256, 0, stream>>>(
        positions, charge, bpk, pref, atom_recip, N);

    ewald_reduce_kernel<<<B, 256, 0, stream>>>(
        charge, atom_real, atom_recip, (float*)d_out, N);
}